// AdversarialFeatureDropout_38903813767348
// MI455X (gfx1250) — compile-verified
//
#include <hip/hip_runtime.h>
#include <hip/hip_bf16.h>

// AdversarialFeatureDropout for MI455X (gfx1250, wave32).
// out[b,s,f] = x[b,s,f] * m[b,f] + a[b,f]
//   m[b,f] = 0 if f dropped (perm2 positions under r-thresholds) or mimicked,
//            1 otherwise
//   a[b,f] = benign_means[y[b], feat_mimic[b]] if mimicked feature, else 0
// Pure HBM-streaming kernel: 512 MiB traffic => ~23 us at 23.3 TB/s.

typedef __attribute__((ext_vector_type(4))) float v4f;

#define B_TOT  2048
#define S_LEN  128
#define F_DIM  256
#define N_DROP 200
#define ROWS_PER_CHUNK 32   // S_LEN / gridDim.y(=4)

__global__ __launch_bounds__(256)
void adv_feat_dropout_kernel(const float* __restrict__ x,
                             const float* __restrict__ benign_means,
                             const float* __restrict__ r,
                             const int*   __restrict__ y,
                             const int*   __restrict__ perm2,
                             const int*   __restrict__ feat_mimic,
                             float*       __restrict__ out) {
    const int b = blockIdx.x;       // sample
    const int t = threadIdx.x;      // 0..255

    // ---- per-sample (mask, add) tables in LDS: one feature per thread ----
    __shared__ v4f smask4[F_DIM / 4];
    __shared__ v4f sadd4[F_DIM / 4];

    const float rv = r[b];
    const int   yb = y[b];
    const bool drop_any = rv < 0.45f;                       // P_SINGLE+P_DOUBLE
    const bool drop_two = rv < 0.15f;                       // P_DOUBLE
    const bool mimic    = (rv >= 0.45f) && (rv < 0.55f) && (yb >= 0) && (yb < 10);
    const int p0  = perm2[2 * b + 0];
    const int p1  = perm2[2 * b + 1];
    const int fmv = feat_mimic[b];

    float m = 1.0f;
    float a = 0.0f;
    if (t < N_DROP) {
        if ((drop_any && t == p0) || (drop_two && t == p1)) m = 0.0f;
    }
    if (mimic && t == fmv) {
        // replaced feature: x*0 + mean  (inputs are finite => exact)
        m = 0.0f;
        a = benign_means[yb * F_DIM + fmv];
    }
    ((float*)smask4)[t] = m;
    ((float*)sadd4)[t]  = a;
    __syncthreads();

    // ---- streaming phase: each thread owns one float4 feature-column ----
    const int col4 = t & 63;                 // float4 column within F (0..63)
    const int rbase = blockIdx.y * ROWS_PER_CHUNK + (t >> 6);   // starting row

    const v4f m4 = smask4[col4];
    const v4f a4 = sadd4[col4];

    const size_t base4 = (size_t)b * (S_LEN * F_DIM / 4);
    const v4f* __restrict__ xb = (const v4f*)x + base4;
    v4f*       __restrict__ ob = (v4f*)out + base4;

#pragma unroll
    for (int i = 0; i < ROWS_PER_CHUNK / 4; ++i) {
        const size_t idx = (size_t)(rbase + 4 * i) * (F_DIM / 4) + col4;
        v4f v = __builtin_nontemporal_load(xb + idx);   // global_load_b128 TH_NT
        v = v * m4 + a4;                                // 4x v_fma_f32
        __builtin_nontemporal_store(v, ob + idx);       // global_store_b128 TH_NT
    }
}

extern "C" void kernel_launch(void* const* d_in, const int* in_sizes, int n_in,
                              void* d_out, int out_size, void* d_ws, size_t ws_size,
                              hipStream_t stream) {
    const float* x  = (const float*)d_in[0];   // (B,S,F) fp32
    const float* bm = (const float*)d_in[1];   // (10,F)  fp32
    const float* r  = (const float*)d_in[2];   // (B,)    fp32
    const int*   y  = (const int*)d_in[3];     // (B,)    i32
    const int*   p2 = (const int*)d_in[4];     // (B,2)   i32
    const int*   fm = (const int*)d_in[5];     // (B,)    i32
    float* out = (float*)d_out;

    const int B = in_sizes[2];                 // 2048
    dim3 grid(B, S_LEN / ROWS_PER_CHUNK, 1);   // (2048, 4)
    adv_feat_dropout_kernel<<<grid, 256, 0, stream>>>(x, bm, r, y, p2, fm, out);
}